// GraphConvolution_22144851378250
// MI455X (gfx1250) — compile-verified
//
#include <hip/hip_runtime.h>

typedef float v2f __attribute__((ext_vector_type(2)));
typedef float v8f __attribute__((ext_vector_type(8)));

#define FEAT 128

// ---------------------------------------------------------------- zero fill
__global__ void zero_f32(float* __restrict__ p, int n) {
  int i = blockIdx.x * blockDim.x + threadIdx.x;
  if (i < n) p[i] = 0.0f;
}

// ---------------------------------------------------------------- GEMM (f32 WMMA)
// S[N,128] = X[N,128] @ W[128,128], all row-major.
// One wave computes one 16x16 output tile; 8 waves/block cover N-dim of W.
// A fragment (16x4 f32): lanes 0-15 -> K=0(,1), lanes 16-31 -> K=2(,3).
__global__ __launch_bounds__(256) void gemm_wmma_f32(
    const float* __restrict__ X, const float* __restrict__ W,
    float* __restrict__ S, int N) {
  const int lane = threadIdx.x & 31;
  const int wave = threadIdx.x >> 5;   // col tile 0..7
  const int half = lane >> 4;          // 0 or 1
  const int l16  = lane & 15;
  const int m0 = blockIdx.x * 16;
  const int n0 = wave * 16;

  int mrow = m0 + l16;
  if (mrow >= N) mrow = N - 1;  // clamp tail so EXEC stays all-ones for WMMA

  const float* xrow = X + (size_t)mrow * FEAT;
  v8f c = {};
  for (int k = 0; k < FEAT; k += 4) {
    const int ka = k + 2 * half;
    v2f a, b;
    a.x = xrow[ka];
    a.y = xrow[ka + 1];
    b.x = W[(size_t)ka       * FEAT + n0 + l16];
    b.y = W[(size_t)(ka + 1) * FEAT + n0 + l16];
    // 8 args: (neg_a, A, neg_b, B, c_mod, C, reuse_a, reuse_b)
    c = __builtin_amdgcn_wmma_f32_16x16x4_f32(false, a, false, b,
                                              (short)0, c, false, false);
  }
  // C/D layout: VGPR r, lanes 0-15 -> M=r, lanes 16-31 -> M=r+8; N = lane%16
  float* srow = S + (size_t)(m0 + 8 * half) * FEAT + n0 + l16;
  if (m0 + 16 <= N) {
    // fast path: full tile, no guards
#pragma unroll
    for (int r = 0; r < 8; ++r) srow[(size_t)r * FEAT] = c[r];
  } else {
#pragma unroll
    for (int r = 0; r < 8; ++r) {
      if (m0 + r + 8 * half < N) srow[(size_t)r * FEAT] = c[r];
    }
  }
}

// ---------------------------------------------------------------- degree
__global__ void degree_kernel(const int* __restrict__ ei, float* __restrict__ deg,
                              int E) {
  int e = blockIdx.x * blockDim.x + threadIdx.x;
  if (e < E) unsafeAtomicAdd(&deg[ei[e]], 1.0f);  // ei[0:E] == row
}

__global__ void dinv_kernel(float* __restrict__ deg,
                            const int* __restrict__ need_norm, int N) {
  int i = blockIdx.x * blockDim.x + threadIdx.x;
  if (i < N) {
    float d = deg[i] + 1.0f;                 // +1 self loop, always > 0
    deg[i] = (*need_norm) ? rsqrtf(d) : 1.0f;
  }
}

// ---------------------------------------------------------------- edge scatter
// agg[row] += support[col] * dinv[col]; one wave per edge, float4 per lane.
__global__ __launch_bounds__(256) void edge_agg(
    const int* __restrict__ ei, const float* __restrict__ S,
    const float* __restrict__ dinv, float* __restrict__ agg, int E) {
  const int lane = threadIdx.x & 31;
  const int e = blockIdx.x * 8 + (threadIdx.x >> 5);
  if (e >= E) return;
  const int row = ei[e];
  const int col = ei[(size_t)E + e];
  const float s = dinv[col];
  const float4 v = *(const float4*)(S + (size_t)col * FEAT + lane * 4);
  float* dst = agg + (size_t)row * FEAT + lane * 4;
  unsafeAtomicAdd(dst + 0, v.x * s);
  unsafeAtomicAdd(dst + 1, v.y * s);
  unsafeAtomicAdd(dst + 2, v.z * s);
  unsafeAtomicAdd(dst + 3, v.w * s);
}

// ---------------------------------------------------------------- finalize
// out = dinv*agg + dinv^2*support + bias  (agg already lives in `out`)
__global__ void finalize_kernel(float* __restrict__ out,
                                const float* __restrict__ S,
                                const float* __restrict__ dinv,
                                const float* __restrict__ bias, int N) {
  int i = blockIdx.x * blockDim.x + threadIdx.x;  // one float4 each
  if (i >= N * (FEAT / 4)) return;
  const int node = i >> 5;           // FEAT/4 == 32
  const int f = (i & 31) * 4;
  const float di = dinv[node];
  const float di2 = di * di;
  float4 a = *(float4*)(out + (size_t)node * FEAT + f);
  const float4 sv = *(const float4*)(S + (size_t)node * FEAT + f);
  const float4 b = *(const float4*)(bias + f);
  a.x = di * a.x + di2 * sv.x + b.x;
  a.y = di * a.y + di2 * sv.y + b.y;
  a.z = di * a.z + di2 * sv.z + b.z;
  a.w = di * a.w + di2 * sv.w + b.w;
  *(float4*)(out + (size_t)node * FEAT + f) = a;
}

// ---------------------------------------------------------------- launch
extern "C" void kernel_launch(void* const* d_in, const int* in_sizes, int n_in,
                              void* d_out, int out_size, void* d_ws, size_t ws_size,
                              hipStream_t stream) {
  const float* X = (const float*)d_in[0];
  const float* W = (const float*)d_in[1];
  const float* bias = (const float*)d_in[2];
  const int* ei = (const int*)d_in[3];
  const int* need_norm = (const int*)d_in[4];

  const int N = in_sizes[0] / FEAT;   // 100000
  const int E = in_sizes[3] / 2;      // 1600000

  float* out = (float*)d_out;                      // doubles as agg accumulator
  float* S = (float*)d_ws;                         // support: N*FEAT floats
  float* dinv = S + (size_t)N * FEAT;              // deg -> dinv: N floats

  // 1) zero agg (= d_out) and deg
  {
    int n = N * FEAT;
    zero_f32<<<(n + 255) / 256, 256, 0, stream>>>(out, n);
    zero_f32<<<(N + 255) / 256, 256, 0, stream>>>(dinv, N);
  }
  // 2) support = X @ W via f32 WMMA
  gemm_wmma_f32<<<(N + 15) / 16, 256, 0, stream>>>(X, W, S, N);
  // 3) degree + normalization factor
  degree_kernel<<<(E + 255) / 256, 256, 0, stream>>>(ei, dinv, E);
  dinv_kernel<<<(N + 255) / 256, 256, 0, stream>>>(dinv, need_norm, N);
  // 4) scatter messages into agg (L2-resident atomics)
  edge_agg<<<(E + 7) / 8, 256, 0, stream>>>(ei, S, dinv, out, E);
  // 5) combine with self term and bias, in place
  {
    int n4 = N * (FEAT / 4);
    finalize_kernel<<<(n4 + 255) / 256, 256, 0, stream>>>(out, S, dinv, bias, N);
  }
}